// DivLoss_25632364822594
// MI455X (gfx1250) — compile-verified
//
#include <hip/hip_runtime.h>

#define TAU 0.85f

typedef float v2f __attribute__((ext_vector_type(2)));
typedef float v8f __attribute__((ext_vector_type(8)));

__device__ __forceinline__ float sum8(v8f c) {
    // fixed association -> deterministic
    return (((c[0] + c[1]) + (c[2] + c[3])) + ((c[4] + c[5]) + (c[6] + c[7])));
}

// One workgroup (256 threads = 8 wave32) per row of 4096 elements.
__global__ __launch_bounds__(256) void divloss_row_kernel(
    const float* __restrict__ p, const int* __restrict__ labels,
    float* __restrict__ row_loss, int L)
{
    const int row  = blockIdx.x;
    const int t    = threadIdx.x;
    const int lane = t & 31;
    const int wave = t >> 5;
    const long long base = (long long)row * L + (long long)t * 16;

    // ---- load 16 labels with 4x b128 loads; keep everything in registers ----
    const int4* lab4 = (const int4*)(labels + base);
    int lv[16];
#pragma unroll
    for (int i = 0; i < 4; ++i) {
        int4 v = lab4[i];
        lv[4*i+0] = v.x; lv[4*i+1] = v.y; lv[4*i+2] = v.z; lv[4*i+3] = v.w;
    }

    // thread-local inclusive cumsum (stays in regs; labels never re-read)
    int cum[16]; int run = 0;
#pragma unroll
    for (int j = 0; j < 16; ++j) { run += lv[j]; cum[j] = run; }

    // wave32 inclusive scan of per-thread totals
    int sc = run;
#pragma unroll
    for (int d = 1; d < 32; d <<= 1) {
        int n = __shfl_up(sc, d, 32);
        if (lane >= d) sc += n;
    }

    __shared__ int sWave[8];
    if (lane == 31) sWave[wave] = sc;
    __syncthreads();
    int waveOff = 0, T = 0;
#pragma unroll
    for (int w = 0; w < 8; ++w) {
        int v = sWave[w];
        if (w < wave) waveOff += v;
        T += v;
    }
    const int   excl = waveOff + (sc - run);   // exclusive prefix for this thread
    const float Tf   = (float)T;

    // ---- single fused pass: f1 -> s -> exp -> three weighted partial sums ----
    const float4* p4 = (const float4*)(p + base);
    float Esum = 0.f, Asum = 0.f, Psum = 0.f;
#pragma unroll
    for (int i = 0; i < 4; ++i) {
        float4 pv = p4[i];
        float pl[4] = {pv.x, pv.y, pv.z, pv.w};
#pragma unroll
        for (int k = 0; k < 4; ++k) {
            int   j   = 4*i + k;
            int   c   = excl + cum[j];
            float pos = (float)(t * 16 + j + 1);
            // f1 = 2c/(T+pos) when c>0 else 0 ; s = f1/TAU  (s in [0, 1.18])
            float s = (c > 0) ? ((2.0f / TAU) * (float)c / (Tf + pos)) : 0.0f;
            float e  = __expf(s);
            float lp = __logf(pl[k]);
            Esum += e;
            Asum += e * s;
            Psum += e * lp;
        }
    }

    // ---- intra-wave reduction on the matrix pipe: V_WMMA_F32_16X16X4_F32 ----
    // A (16x4 f32): lane l<16 holds (A[l,0],A[l,1]); lane l>=16 holds (A[l-16,2],A[l-16,3]).
    // B (4x16 f32): column 0 selects k={0,2}  -> C[m,0] = Esum_m + Esum_{m+16}
    //               column 1 selects k={1,3}  -> C[m,1] = Asum_m + Asum_{m+16}
    v2f a1; a1.x = Esum; a1.y = Asum;
    v2f a2; a2.x = Psum; a2.y = 0.0f;
    v2f b;
    b.x = (lane == 0 || lane == 16) ? 1.0f : 0.0f;   // B[0,0]=B[2,0]=1
    b.y = (lane == 1 || lane == 17) ? 1.0f : 0.0f;   // B[1,1]=B[3,1]=1
    v8f c1 = {}; v8f c2 = {};
    c1 = __builtin_amdgcn_wmma_f32_16x16x4_f32(false, a1, false, b, (short)0, c1, false, false);
    c2 = __builtin_amdgcn_wmma_f32_16x16x4_f32(false, a2, false, b, (short)0, c2, false, false);

    // Column n of C lives in lane n (rows 0-7) and lane n+16 (rows 8-15).
    float l1 = sum8(c1), l2 = sum8(c2);
    float o1 = __shfl(l1, lane + 16, 32);   // meaningful for lanes 0..15
    float o2 = __shfl(l2, lane + 16, 32);
    float vEA = l1 + o1;                    // lane0: wave E, lane1: wave A
    float vP  = l2 + o2;                    // lane0: wave P

    __shared__ float sE[8], sA[8], sP[8];
    if (lane == 0) { sE[wave] = vEA; sP[wave] = vP; }
    if (lane == 1) { sA[wave] = vEA; }
    __syncthreads();

    if (t == 0) {
        float E = 0.f, A = 0.f, P = 0.f;
#pragma unroll
        for (int w = 0; w < 8; ++w) { E += sE[w]; A += sA[w]; P += sP[w]; }
        // sum q*(log q - log p) = (A - P)/E - log(E)
        row_loss[row] = (A - P) / E - __logf(E);
    }
}

// Deterministic fixed-order final reduction of per-row losses.
__global__ __launch_bounds__(256) void divloss_final_kernel(
    const float* __restrict__ rl, float* __restrict__ out, int B)
{
    __shared__ float red[256];
    int t = threadIdx.x;
    float v = 0.f;
    for (int i = t; i < B; i += 256) v += rl[i];  // fixed order per thread
    red[t] = v;
    __syncthreads();
    for (int s = 128; s > 0; s >>= 1) {
        if (t < s) red[t] += red[t + s];
        __syncthreads();
    }
    if (t == 0) out[0] = red[0] / (float)B;
}

extern "C" void kernel_launch(void* const* d_in, const int* in_sizes, int n_in,
                              void* d_out, int out_size, void* d_ws, size_t ws_size,
                              hipStream_t stream)
{
    const float* p      = (const float*)d_in[0]; // (B, L, 1) f32
    const int*   labels = (const int*)d_in[1];   // (B, L)   i32
    const int L = 4096;
    const int B = in_sizes[1] / L;

    float* rl = (float*)d_ws;                    // B floats of scratch
    divloss_row_kernel<<<B, 256, 0, stream>>>(p, labels, rl, L);
    divloss_final_kernel<<<1, 256, 0, stream>>>(rl, (float*)d_out, B);
}